// RWKV_RNN_67680094650635
// MI455X (gfx1250) — compile-verified
//
#include <hip/hip_runtime.h>
#include <math.h>

// RWKV-4 single-token inference for MI455X (gfx1250).
// Bandwidth-bound mat-vec chain; mat-vecs via V_WMMA_F32_16X16X4_F32 with
// ping-pong software-pipelined NT loads, deep prefetch, K-split for occupancy.

#define LYR 24
#define EMBD 1024
#define HID 4096
#define VOC 50257
#define TPB 256
#define NWAVES 8   // 256 threads / wave32

typedef float v2f __attribute__((ext_vector_type(2)));
typedef float v8f __attribute__((ext_vector_type(8)));

// ---------------- block-wide sum over 256 threads ----------------
__device__ __forceinline__ float block_sum256(float v, float* red) {
  int t = threadIdx.x;
  red[t] = v;
  __syncthreads();
#pragma unroll
  for (int s = 128; s > 0; s >>= 1) {
    if (t < s) red[t] += red[t + s];
    __syncthreads();
  }
  float r = red[0];
  __syncthreads();
  return r;
}

// ---------------- x = LN(emb[token], w, b) ----------------
__global__ void __launch_bounds__(TPB) embed_ln_kernel(
    const int* __restrict__ token, const float* __restrict__ emb,
    const float* __restrict__ w, const float* __restrict__ b,
    float* __restrict__ xout) {
  __shared__ float red[TPB];
  const float* row = emb + (size_t)(*token) * EMBD;
  float s = 0.f;
  for (int i = threadIdx.x; i < EMBD; i += TPB) s += row[i];
  float mean = block_sum256(s, red) * (1.0f / EMBD);
  float vs = 0.f;
  for (int i = threadIdx.x; i < EMBD; i += TPB) { float d = row[i] - mean; vs += d * d; }
  float var = block_sum256(vs, red) * (1.0f / EMBD);
  float inv = rsqrtf(var + 1e-5f);
  for (int i = threadIdx.x; i < EMBD; i += TPB)
    xout[i] = (row[i] - mean) * inv * w[i] + b[i];
}

// ---------------- final LN ----------------
__global__ void __launch_bounds__(TPB) final_ln_kernel(
    const float* __restrict__ x, const float* __restrict__ w,
    const float* __restrict__ b, float* __restrict__ xout) {
  __shared__ float red[TPB];
  float s = 0.f;
  for (int i = threadIdx.x; i < EMBD; i += TPB) s += x[i];
  float mean = block_sum256(s, red) * (1.0f / EMBD);
  float vs = 0.f;
  for (int i = threadIdx.x; i < EMBD; i += TPB) { float d = x[i] - mean; vs += d * d; }
  float var = block_sum256(vs, red) * (1.0f / EMBD);
  float inv = rsqrtf(var + 1e-5f);
  for (int i = threadIdx.x; i < EMBD; i += TPB)
    xout[i] = (x[i] - mean) * inv * w[i] + b[i];
}

// ---------------- time-mix prologue: xa = LN(x); blend with s_att ----------------
__global__ void __launch_bounds__(TPB) time_mix_pre_kernel(
    const float* __restrict__ x, const float* __restrict__ w, const float* __restrict__ b,
    const float* __restrict__ mk, const float* __restrict__ mv, const float* __restrict__ mr,
    const float* __restrict__ s_att,  // old state row 1
    float* __restrict__ xa_state,     // new state row 1
    float* __restrict__ xk, float* __restrict__ xv, float* __restrict__ xr) {
  __shared__ float red[TPB];
  float s = 0.f;
  for (int i = threadIdx.x; i < EMBD; i += TPB) s += x[i];
  float mean = block_sum256(s, red) * (1.0f / EMBD);
  float vs = 0.f;
  for (int i = threadIdx.x; i < EMBD; i += TPB) { float d = x[i] - mean; vs += d * d; }
  float var = block_sum256(vs, red) * (1.0f / EMBD);
  float inv = rsqrtf(var + 1e-5f);
  for (int i = threadIdx.x; i < EMBD; i += TPB) {
    float xa = (x[i] - mean) * inv * w[i] + b[i];
    xa_state[i] = xa;
    float sa = s_att[i];
    float a = mk[i]; xk[i] = xa * a + sa * (1.f - a);
    float v = mv[i]; xv[i] = xa * v + sa * (1.f - v);
    float r = mr[i]; xr[i] = xa * r + sa * (1.f - r);
  }
}

// ---------------- channel-mix prologue: x += atty; xc = LN(x); blend with s_ffn ----------------
__global__ void __launch_bounds__(TPB) channel_mix_pre_kernel(
    float* __restrict__ x, const float* __restrict__ atty,
    const float* __restrict__ w, const float* __restrict__ b,
    const float* __restrict__ fmk, const float* __restrict__ fmr,
    const float* __restrict__ s_ffn,  // old state row 0
    float* __restrict__ xc_state,     // new state row 0
    float* __restrict__ xk2, float* __restrict__ xr2) {
  __shared__ float red[TPB];
  float s = 0.f;
  for (int i = threadIdx.x; i < EMBD; i += TPB) {
    float t = x[i] + atty[i];
    x[i] = t;
    s += t;
  }
  float mean = block_sum256(s, red) * (1.0f / EMBD);  // syncthreads makes x[] stores visible
  float vs = 0.f;
  for (int i = threadIdx.x; i < EMBD; i += TPB) { float d = x[i] - mean; vs += d * d; }
  float var = block_sum256(vs, red) * (1.0f / EMBD);
  float inv = rsqrtf(var + 1e-5f);
  for (int i = threadIdx.x; i < EMBD; i += TPB) {
    float xc = (x[i] - mean) * inv * w[i] + b[i];
    xc_state[i] = xc;
    float sf = s_ffn[i];
    float a = fmk[i]; xk2[i] = xc * a + sf * (1.f - a);
    float r = fmr[i]; xr2[i] = xc * r + sf * (1.f - r);
  }
}

// ---------------- WKV state update (elementwise) ----------------
__global__ void wkv_kernel(
    const float* __restrict__ k, const float* __restrict__ v, const float* __restrict__ r,
    const float* __restrict__ tf, const float* __restrict__ td,
    const float* __restrict__ aa_in, const float* __restrict__ bb_in, const float* __restrict__ pp_in,
    float* __restrict__ aa_out, float* __restrict__ bb_out, float* __restrict__ pp_out,
    float* __restrict__ rwkv) {
  for (int i = threadIdx.x + blockIdx.x * blockDim.x; i < EMBD; i += blockDim.x * gridDim.x) {
    float kk = k[i], vv = v[i];
    float aa = aa_in[i], bb = bb_in[i], pp = pp_in[i];
    float ww = tf[i] + kk;
    float qq = fmaxf(pp, ww);
    float e1 = expf(pp - qq), e2 = expf(ww - qq);
    float wkv = (e1 * aa + e2 * vv) / (e1 * bb + e2);
    rwkv[i] = r[i] * wkv;
    float ww2 = pp + td[i];
    float qq2 = fmaxf(ww2, kk);
    float e1b = expf(ww2 - qq2), e2b = expf(kk - qq2);
    aa_out[i] = e1b * aa + e2b * vv;
    bb_out[i] = e1b * bb + e2b;
    pp_out[i] = qq2;
  }
}

// ---------------- x += r2 * fv ----------------
__global__ void channel_mix_post_kernel(float* __restrict__ x,
                                        const float* __restrict__ r2,
                                        const float* __restrict__ fv) {
  int i = threadIdx.x + blockIdx.x * blockDim.x;
  if (i < EMBD) x[i] += r2[i] * fv[i];
}

// ---------------- mat-vec via V_WMMA_F32_16X16X4_F32 ----------------
// y[M] = act(W[M,K] @ x[K]).  grid = (ceil(M/16), ksplit).
// Block owns 16 output rows and a K-chunk; its 8 waves split the chunk.
// Ping-pong pipeline (unroll x2, two register sets): each set is written by
// its own NT b64 loads and consumed directly by its WMMAs -- no rotation
// copies, waits land next to the consuming WMMAs, ~2KB per wave in flight.
// Prefetch runs 512B ahead along each row stream (speculative, OOB-safe).
// A (16x4 f32): lane-half h holds K=2h,2h+1 -> broadcast of x.
// B (4x16 f32): lane n / half h holds W[row n, k+2h .. +1].
// All rows of D identical; D VGPR0 lanes 0..15 carry the 16 outputs.
// act: 0 = none, 1 = sigmoid, 2 = relu^2 (applied here only when ksplit==1)
__global__ void __launch_bounds__(TPB) matvec16_wmma_kernel(
    const float* __restrict__ W, const float* __restrict__ xin,
    float* __restrict__ y, float* __restrict__ part,
    int M, int K, int act) {
  extern __shared__ float smem[];
  const int ksplit = gridDim.y;
  const int kchunk = K / ksplit;        // multiple of 256 in all uses
  const int kbase = blockIdx.y * kchunk;

  float* sx = smem;                      // kchunk floats of x
  float* spart = smem + kchunk;          // 16*NWAVES partials
  for (int i = threadIdx.x; i < kchunk; i += TPB) sx[i] = xin[kbase + i];
  __syncthreads();

  const int tid = threadIdx.x;
  const int wave = tid >> 5;
  const int lane = tid & 31;
  const int n = lane & 15;
  const int h = lane >> 4;  // 0 or 1
  int row = blockIdx.x * 16 + n;
  if (row >= M) row = M - 1;  // clamp: safe loads, tail rows not stored
  const float* wrow = W + (size_t)row * K + kbase;
  const float* wp = wrow + 2 * h;

  const int kslice = kchunk / NWAVES;    // multiple of 32 in all uses
  const int k0 = wave * kslice;
  const int k1 = k0 + kslice;
  const v2f* sx2 = reinterpret_cast<const v2f*>(sx);

  v8f c0 = {0.f, 0.f, 0.f, 0.f, 0.f, 0.f, 0.f, 0.f};
  v8f c1 = {0.f, 0.f, 0.f, 0.f, 0.f, 0.f, 0.f, 0.f};

  // set-A prologue: load first half-iteration @ k0
  v2f b0 = __builtin_nontemporal_load(reinterpret_cast<const v2f*>(wp + k0));
  v2f b1 = __builtin_nontemporal_load(reinterpret_cast<const v2f*>(wp + k0 + 4));
  v2f b2 = __builtin_nontemporal_load(reinterpret_cast<const v2f*>(wp + k0 + 8));
  v2f b3 = __builtin_nontemporal_load(reinterpret_cast<const v2f*>(wp + k0 + 12));
  int ai = (k0 >> 1) + h;
  v2f a0 = sx2[ai];
  v2f a1 = sx2[ai + 2];
  v2f a2 = sx2[ai + 4];
  v2f a3 = sx2[ai + 6];

  for (int k = k0; k < k1; k += 32) {
    // ---- issue set-B loads @ k+16 (always in range: kslice % 32 == 0) ----
    const int kB = k + 16;
    v2f d0 = __builtin_nontemporal_load(reinterpret_cast<const v2f*>(wp + kB));
    v2f d1 = __builtin_nontemporal_load(reinterpret_cast<const v2f*>(wp + kB + 4));
    v2f d2 = __builtin_nontemporal_load(reinterpret_cast<const v2f*>(wp + kB + 8));
    v2f d3 = __builtin_nontemporal_load(reinterpret_cast<const v2f*>(wp + kB + 12));
    const int bi = (kB >> 1) + h;
    v2f e0 = sx2[bi];
    v2f e1 = sx2[bi + 2];
    v2f e2 = sx2[bi + 4];
    v2f e3 = sx2[bi + 6];
    __builtin_prefetch(wrow + k + 128, 0, 1);  // 512B ahead
    // ---- consume set-A ----
    c0 = __builtin_amdgcn_wmma_f32_16x16x4_f32(false, a0, false, b0, (short)0, c0, false, false);
    c1 = __builtin_amdgcn_wmma_f32_16x16x4_f32(false, a1, false, b1, (short)0, c1, false, false);
    c0 = __builtin_amdgcn_wmma_f32_16x16x4_f32(false, a2, false, b2, (short)0, c0, false, false);
    c1 = __builtin_amdgcn_wmma_f32_16x16x4_f32(false, a3, false, b3, (short)0, c1, false, false);

    // ---- issue set-A loads @ k+32 (uniform clamp; last trip reloads k0) ----
    const int kA = (k + 32 < k1) ? (k + 32) : k0;
    b0 = __builtin_nontemporal_load(reinterpret_cast<const v2f*>(wp + kA));
    b1 = __builtin_nontemporal_load(reinterpret_cast<const v2f*>(wp + kA + 4));
    b2 = __builtin_nontemporal_load(reinterpret_cast<const v2f*>(wp + kA + 8));
    b3 = __builtin_nontemporal_load(reinterpret_cast<const v2f*>(wp + kA + 12));
    const int ci = (kA >> 1) + h;
    a0 = sx2[ci];
    a1 = sx2[ci + 2];
    a2 = sx2[ci + 4];
    a3 = sx2[ci + 6];
    __builtin_prefetch(wrow + kB + 128, 0, 1);  // 512B ahead
    // ---- consume set-B ----
    c0 = __builtin_amdgcn_wmma_f32_16x16x4_f32(false, e0, false, d0, (short)0, c0, false, false);
    c1 = __builtin_amdgcn_wmma_f32_16x16x4_f32(false, e1, false, d1, (short)0, c1, false, false);
    c0 = __builtin_amdgcn_wmma_f32_16x16x4_f32(false, e2, false, d2, (short)0, c0, false, false);
    c1 = __builtin_amdgcn_wmma_f32_16x16x4_f32(false, e3, false, d3, (short)0, c1, false, false);
  }
  float partial = c0[0] + c1[0];
  if (lane < 16) spart[wave * 16 + n] = partial;
  __syncthreads();
  if (tid < 16) {
    float s = 0.f;
#pragma unroll
    for (int wv = 0; wv < NWAVES; ++wv) s += spart[wv * 16 + tid];
    int orow = blockIdx.x * 16 + tid;
    if (orow < M) {
      if (ksplit == 1) {
        if (act == 1) s = 1.0f / (1.0f + expf(-s));
        else if (act == 2) { s = fmaxf(s, 0.f); s = s * s; }
        y[orow] = s;
      } else {
        part[(size_t)blockIdx.y * M + orow] = s;
      }
    }
  }
}

// ---------------- combine K-split partials + activation ----------------
__global__ void reduce_act_kernel(const float* __restrict__ part, float* __restrict__ y,
                                  int M, int ksplit, int act) {
  int i = threadIdx.x + blockIdx.x * blockDim.x;
  if (i >= M) return;
  float s = 0.f;
  for (int ks = 0; ks < ksplit; ++ks) s += part[(size_t)ks * M + i];
  if (act == 1) s = 1.0f / (1.0f + expf(-s));
  else if (act == 2) { s = fmaxf(s, 0.f); s = s * s; }
  y[i] = s;
}

static inline void launch_matvec(const float* W, const float* xin, float* y, float* part,
                                 int M, int K, int act, hipStream_t s) {
  const int ksplit = (M <= 2048) ? 4 : 1;  // boost wave count for small-M matvecs
  const int blocksM = (M + 15) / 16;
  const int kchunk = K / ksplit;
  size_t shmem = (size_t)(kchunk + 16 * NWAVES) * sizeof(float);
  dim3 grid(blocksM, ksplit);
  matvec16_wmma_kernel<<<grid, TPB, shmem, s>>>(W, xin, y, part, M, K, act);
  if (ksplit > 1)
    reduce_act_kernel<<<(M + TPB - 1) / TPB, TPB, 0, s>>>(part, y, M, ksplit, act);
}

extern "C" void kernel_launch(void* const* d_in, const int* in_sizes, int n_in,
                              void* d_out, int out_size, void* d_ws, size_t ws_size,
                              hipStream_t stream) {
  (void)in_sizes; (void)n_in; (void)out_size; (void)ws_size;

  const int*   token  = (const int*)d_in[0];
  const float* state  = (const float*)d_in[1];
  const float* emb    = (const float*)d_in[2];
  const float* ln0w   = (const float*)d_in[3];
  const float* ln0b   = (const float*)d_in[4];
  const float* ln1w   = (const float*)d_in[5];
  const float* ln1b   = (const float*)d_in[6];
  const float* ln2w   = (const float*)d_in[7];
  const float* ln2b   = (const float*)d_in[8];
  const float* tmk    = (const float*)d_in[9];
  const float* tmv    = (const float*)d_in[10];
  const float* tmr    = (const float*)d_in[11];
  const float* tfirst = (const float*)d_in[12];
  const float* tdecay = (const float*)d_in[13];
  const float* attkw  = (const float*)d_in[14];
  const float* attvw  = (const float*)d_in[15];
  const float* attrw  = (const float*)d_in[16];
  const float* attow  = (const float*)d_in[17];
  const float* ftmk   = (const float*)d_in[18];
  const float* ftmr   = (const float*)d_in[19];
  const float* ffnkw  = (const float*)d_in[20];
  const float* ffnvw  = (const float*)d_in[21];
  const float* ffnrw  = (const float*)d_in[22];
  const float* lnoW   = (const float*)d_in[23];
  const float* lnoB   = (const float*)d_in[24];
  const float* head   = (const float*)d_in[25];

  float* logits = (float*)d_out;              // [VOC]
  float* ostate = (float*)d_out + VOC;        // [LYR,5,EMBD]

  // workspace layout (floats)
  float* ws   = (float*)d_ws;
  float* x    = ws + 0 * EMBD;
  float* xk   = ws + 1 * EMBD;
  float* xv   = ws + 2 * EMBD;
  float* xr   = ws + 3 * EMBD;
  float* kbuf = ws + 4 * EMBD;
  float* vbuf = ws + 5 * EMBD;
  float* rbuf = ws + 6 * EMBD;   // sigmoid already applied
  float* rwkv = ws + 7 * EMBD;
  float* atty = ws + 8 * EMBD;
  float* xk2  = ws + 9 * EMBD;
  float* xr2  = ws + 10 * EMBD;
  float* r2   = ws + 11 * EMBD;  // sigmoid already applied
  float* fv   = ws + 12 * EMBD;
  float* kkb  = ws + 13 * EMBD;             // HID floats (relu^2 applied)
  float* part = ws + 13 * EMBD + HID;       // 4*EMBD K-split partials

  embed_ln_kernel<<<1, TPB, 0, stream>>>(token, emb, ln0w, ln0b, x);

  for (int l = 0; l < LYR; ++l) {
    const float* sl = state + (size_t)l * 5 * EMBD;   // old state, layer l
    float* ol = ostate + (size_t)l * 5 * EMBD;        // new state, layer l
    const size_t mm = (size_t)l * EMBD * EMBD;

    time_mix_pre_kernel<<<1, TPB, 0, stream>>>(
        x, ln1w + l * EMBD, ln1b + l * EMBD,
        tmk + l * EMBD, tmv + l * EMBD, tmr + l * EMBD,
        sl + 1 * EMBD,            // s_att
        ol + 1 * EMBD,            // state row 1 = xa
        xk, xv, xr);

    launch_matvec(attkw + mm, xk, kbuf, part, EMBD, EMBD, 0, stream);
    launch_matvec(attvw + mm, xv, vbuf, part, EMBD, EMBD, 0, stream);
    launch_matvec(attrw + mm, xr, rbuf, part, EMBD, EMBD, 1, stream);  // sigmoid

    wkv_kernel<<<4, TPB, 0, stream>>>(
        kbuf, vbuf, rbuf, tfirst + l * EMBD, tdecay + l * EMBD,
        sl + 2 * EMBD, sl + 3 * EMBD, sl + 4 * EMBD,   // aa, bb, pp (old)
        ol + 2 * EMBD, ol + 3 * EMBD, ol + 4 * EMBD,   // aa2, bb2, qq2 (new)
        rwkv);

    launch_matvec(attow + mm, rwkv, atty, part, EMBD, EMBD, 0, stream);

    channel_mix_pre_kernel<<<1, TPB, 0, stream>>>(
        x, atty, ln2w + l * EMBD, ln2b + l * EMBD,
        ftmk + l * EMBD, ftmr + l * EMBD,
        sl + 0 * EMBD,            // s_ffn
        ol + 0 * EMBD,            // state row 0 = xc
        xk2, xr2);

    launch_matvec(ffnkw + (size_t)l * HID * EMBD, xk2, kkb, part, HID, EMBD, 2, stream); // relu^2
    launch_matvec(ffnrw + mm, xr2, r2, part, EMBD, EMBD, 1, stream);                     // sigmoid
    launch_matvec(ffnvw + (size_t)l * EMBD * HID, kkb, fv, part, EMBD, HID, 0, stream);

    channel_mix_post_kernel<<<4, TPB, 0, stream>>>(x, r2, fv);
  }

  final_ln_kernel<<<1, TPB, 0, stream>>>(x, lnoW, lnoB, xk);  // reuse xk slot
  launch_matvec(head, xk, logits, part, VOC, EMBD, 0, stream);
}